// RGNN_52372831208114
// MI455X (gfx1250) — compile-verified
//
#include <hip/hip_runtime.h>
#include <hip/hip_bf16.h>

#define N_NODES 62
#define F_IN    5
#define HID     400
#define WAVES_PER_BLOCK 8

typedef __attribute__((ext_vector_type(2))) float v2f;
typedef __attribute__((ext_vector_type(8))) float v8f;

__device__ __forceinline__ v8f wmma_f32_16x16x4(v2f a, v2f b, v8f c) {
#if defined(__gfx1250__)
    // D = A(16x4 f32) x B(4x16 f32) + C(16x16 f32); SRC2 may differ from VDST
    return __builtin_amdgcn_wmma_f32_16x16x4_f32(
        /*neg_a=*/false, a, /*neg_b=*/false, b,
        /*c_mod=*/(short)0, c, /*reuse_a=*/false, /*reuse_b=*/false);
#else
    (void)a; (void)b;
    return c;
#endif
}

// ---------------------------------------------------------------------------
// Kernel 1: build A = D^-1/2 W D^-1/2 from lower-tri params, A2 = A*A,
// write A2 zero-padded to [64][64] into workspace. One block, tiny work.
// ---------------------------------------------------------------------------
__global__ void rgnn_prep_kernel(const float* __restrict__ ew_tril,
                                 const int* __restrict__ tril_x,
                                 const int* __restrict__ tril_y,
                                 float* __restrict__ A2g) {
    __shared__ float sW[N_NODES * 64];
    __shared__ float sA[N_NODES * 64];
    __shared__ float sDinv[64];
    const int tid = threadIdx.x;

    for (int i = tid; i < N_NODES * 64; i += 256) sW[i] = 0.0f;
    __syncthreads();

    // 62*63/2 = 1953 unique lower-tri entries (diagonal included once)
    for (int t = tid; t < (N_NODES * (N_NODES + 1)) / 2; t += 256) {
        int x = tril_x[t], y = tril_y[t];
        float w = ew_tril[t];
        sW[x * 64 + y] = w;
        sW[y * 64 + x] = w;   // symmetric fill (diag overwrites itself)
    }
    __syncthreads();

    if (tid < N_NODES) {
        float d = 0.0f;
        for (int j = 0; j < N_NODES; ++j) d += fabsf(sW[tid * 64 + j]);
        sDinv[tid] = (d > 0.0f) ? (1.0f / sqrtf(d)) : 0.0f;
    }
    __syncthreads();

    for (int idx = tid; idx < N_NODES * N_NODES; idx += 256) {
        int i = idx / N_NODES, j = idx % N_NODES;
        sA[i * 64 + j] = sDinv[i] * sW[i * 64 + j] * sDinv[j];
    }
    __syncthreads();

    // A2 padded to 64x64 (rows/cols 62,63 zero so WMMA pad rows give Y=0)
    for (int idx = tid; idx < 64 * 64; idx += 256) {
        int r = idx >> 6, c = idx & 63;
        float s = 0.0f;
        if (r < N_NODES && c < N_NODES) {
            for (int k = 0; k < N_NODES; ++k)
                s += sA[r * 64 + k] * sA[k * 64 + c];
        }
        A2g[idx] = s;
    }
}

// ---------------------------------------------------------------------------
// Kernel 2: fused  Y = A2*X[b];  H = relu(Y*lin_w + b);  pool; out = p*fc + b
// One wave (32 lanes) per batch graph; 8 graphs per block share LDS operands.
// ---------------------------------------------------------------------------
__global__ void __launch_bounds__(256)
rgnn_main_kernel(const float* __restrict__ X,
                 const float* __restrict__ lin_w,
                 const float* __restrict__ lin_b,
                 const float* __restrict__ fc_w,
                 const float* __restrict__ fc_b,
                 const float* __restrict__ A2g,
                 float* __restrict__ out) {
    __shared__ float sA2[64 * 65];                 // padded stride 65 (bank-safe)
    __shared__ float sX[WAVES_PER_BLOCK][64 * 8];  // per-wave X tile, K padded 5->8
    __shared__ float sLin[8 * HID];                // [k][h], k rows 5..7 zero
    __shared__ float sBias[HID];
    __shared__ float sFc[HID * 2];

    const int tid  = threadIdx.x;
    const int wave = tid >> 5;
    const int lane = tid & 31;
    const int b    = blockIdx.x * WAVES_PER_BLOCK + wave;

    // ---- cooperative loads of batch-invariant operands --------------------
    for (int idx = tid; idx < 64 * 64; idx += 256) {
        int r = idx >> 6, c = idx & 63;
        sA2[r * 65 + c] = A2g[idx];
    }
    for (int idx = tid; idx < 8 * HID; idx += 256) {
        int k = idx / HID;
        sLin[idx] = (k < F_IN) ? lin_w[idx] : 0.0f;   // idx = k*HID + h
    }
    for (int idx = tid; idx < HID; idx += 256) sBias[idx] = lin_b[idx];
    for (int idx = tid; idx < HID * 2; idx += 256) sFc[idx] = fc_w[idx];

    // ---- per-wave X staging (zero-padded to [64][8]) ----------------------
    {
        const float* Xb = X + (size_t)b * (N_NODES * F_IN);
        for (int idx = lane; idx < 64 * 8; idx += 32) {
            int i = idx >> 3, k = idx & 7;
            float v = 0.0f;
            if (i < N_NODES && k < F_IN) v = Xb[i * F_IN + k];
            sX[wave][idx] = v;
        }
    }
    __syncthreads();

    // ---- stage 1: Y = A2 * X[b], computed directly in A-fragment layout ---
    // 16x4 f32 A layout: lanes 0-15 rows M, K={0,1}; lanes 16-31 K={2,3}
    const int lh  = lane >> 4;      // lane half
    const int col = lane & 15;
    const int kb  = 2 * lh;         // k base: 0 (lanes<16) or 2 (lanes>=16)

    float y0[4], y1[4], y2[4];      // per M-tile: k = kb, kb+1, kb+4
    #pragma unroll
    for (int m = 0; m < 4; ++m) { y0[m] = 0.f; y1[m] = 0.f; y2[m] = 0.f; }

    const float* sXw = sX[wave];
    for (int i = 0; i < N_NODES; ++i) {
        float xa = sXw[i * 8 + kb];
        float xb = sXw[i * 8 + kb + 1];
        float xc = sXw[i * 8 + kb + 4];   // k>=5 entries are zero padding
        #pragma unroll
        for (int m = 0; m < 4; ++m) {
            float a = sA2[(16 * m + col) * 65 + i];
            y0[m] = fmaf(a, xa, y0[m]);
            y1[m] = fmaf(a, xb, y1[m]);
            y2[m] = fmaf(a, xc, y2[m]);
        }
    }

    v2f afrag0[4], afrag1[4];       // K-steps 0..3 and 4..7
    #pragma unroll
    for (int m = 0; m < 4; ++m) {
        afrag0[m].x = y0[m]; afrag0[m].y = y1[m];
        afrag1[m].x = y2[m]; afrag1[m].y = 0.0f;   // k = kb+5 >= 5 -> zero
    }

    // ---- stage 2: per 16-col HID tile: WMMA(+bias in C) + relu + pool + fc
    float acc0 = 0.0f, acc1 = 0.0f;
    const float rowmask = (lh == 1) ? 0.0f : 1.0f;  // pad rows 62,63 live at
                                                    // m==3, lanes>=16, r=6,7
    for (int j = 0; j < HID / 16; ++j) {            // 25 tiles
        const int colg = j * 16 + col;
        v2f b0, b1;                                  // 4x16 B fragments
        b0.x = sLin[(kb)     * HID + colg];
        b0.y = sLin[(kb + 1) * HID + colg];
        b1.x = sLin[(kb + 4) * HID + colg];          // rows 5..7 zero-padded
        b1.y = sLin[(kb + 5) * HID + colg];

        const float bias = sBias[colg];
        const float f0 = sFc[2 * colg];
        const float f1 = sFc[2 * colg + 1];

        // bias folded into the accumulator init: D = Y*W + bias directly
        v8f cinit;
        #pragma unroll
        for (int r = 0; r < 8; ++r) cinit[r] = bias;

        float msum[4];
        #pragma unroll
        for (int m = 0; m < 4; ++m) {
            v8f c = wmma_f32_16x16x4(afrag0[m], b0, cinit);
            c     = wmma_f32_16x16x4(afrag1[m], b1, c);

            float t[8];
            #pragma unroll
            for (int r = 0; r < 8; ++r) {
                float h = c[r];
                t[r] = (h > 0.0f) ? h : 0.0f;   // single v_max_num_f32 0,h
            }
            if (m == 3) { t[6] *= rowmask; t[7] *= rowmask; } // zero pad rows
            // pairwise tree (dual-issue / v_pk_add friendly)
            float s01 = t[0] + t[1], s23 = t[2] + t[3];
            float s45 = t[4] + t[5], s67 = t[6] + t[7];
            msum[m] = (s01 + s23) + (s45 + s67);
        }
        float psum = (msum[0] + msum[1]) + (msum[2] + msum[3]);
        acc0 = fmaf(psum, f0, acc0);
        acc1 = fmaf(psum, f1, acc1);
    }

    // ---- wave32 reduction across lanes (rows x columns fully covered) -----
    #pragma unroll
    for (int off = 16; off > 0; off >>= 1) {
        acc0 += __shfl_xor(acc0, off, 32);
        acc1 += __shfl_xor(acc1, off, 32);
    }
    if (lane == 0) {
        out[b * 2 + 0] = acc0 + fc_b[0];
        out[b * 2 + 1] = acc1 + fc_b[1];
    }
}

// ---------------------------------------------------------------------------
extern "C" void kernel_launch(void* const* d_in, const int* in_sizes, int n_in,
                              void* d_out, int out_size, void* d_ws, size_t ws_size,
                              hipStream_t stream) {
    (void)in_sizes; (void)n_in; (void)out_size; (void)ws_size;
    const float* X      = (const float*)d_in[0];
    const float* ew     = (const float*)d_in[2];
    const float* lin_w  = (const float*)d_in[3];
    const float* lin_b  = (const float*)d_in[4];
    const float* fc_w   = (const float*)d_in[5];
    const float* fc_b   = (const float*)d_in[6];
    const int*   tril_x = (const int*)d_in[9];
    const int*   tril_y = (const int*)d_in[10];
    float*       out    = (float*)d_out;
    float*       A2g    = (float*)d_ws;            // 64*64 f32 = 16 KB scratch

    rgnn_prep_kernel<<<1, 256, 0, stream>>>(ew, tril_x, tril_y, A2g);
    rgnn_main_kernel<<<4096 / WAVES_PER_BLOCK, 256, 0, stream>>>(
        X, lin_w, lin_b, fc_w, fc_b, A2g, out);
}